// adaPnt_43465069035695
// MI455X (gfx1250) — compile-verified
//
#include <hip/hip_runtime.h>
#include <hip/hip_bf16.h>

typedef __attribute__((ext_vector_type(2))) float v2f;
typedef __attribute__((ext_vector_type(4))) float v4f;
typedef __attribute__((ext_vector_type(8))) float v8f;

#define B_DIM 128
#define RNN 1024
#define ATT 512
#define ROI 1023
#define COLS 1024   // ROI + 1
#define MIN_VALUE -1e8f

// ---------------------------------------------------------------------------
// f32 GEMM via v_wmma_f32_16x16x4_f32.
//   C[M x N] = act(A[M x K] @ B[K x N] + bias[N]),  M fixed at 128.
// Block: 256 threads = 8 waves; wave w owns rows [16w, 16w+16), block owns
// cols [16*blockIdx.x, +16). K stepped by 4 per WMMA.
//
// VGPR layouts (ISA 7.12.2, 32-bit):
//   A 16x4 : lane l (l<16): row=l, K = {0,1}; lane l>=16: row=l-16, K={2,3}
//   B 4x16 : lane l (l<16): col=l, K=v;      lane l>=16: col=l-16, K=v+2
//   C 16x16: vgpr v, lanes 0-15: M=v, N=lane; lanes 16-31: M=v+8, N=lane-16
// ---------------------------------------------------------------------------
__global__ void gemm128_wmma_f32(const float* __restrict__ A,
                                 const float* __restrict__ Bm,
                                 const float* __restrict__ bias,
                                 float* __restrict__ C,
                                 int K, int N, int relu) {
    const int lane = threadIdx.x & 31;
    const int wave = threadIdx.x >> 5;          // 0..7 -> M tile
    const int m0   = wave << 4;
    const int n0   = blockIdx.x << 4;
    const int lr   = lane & 15;
    const int kh   = (lane >> 4) << 1;          // 0 for lanes 0-15, 2 for 16-31

    const float* Arow = A  + (size_t)(m0 + lr) * K + kh;   // consecutive K pair
    const float* Bcol = Bm + (size_t)kh * N + (n0 + lr);   // K rows kh, kh+1

    v8f acc = {};
#pragma unroll 4
    for (int k = 0; k < K; k += 4) {
        const float2 av = *(const float2*)(Arow + k);
        v2f a, b;
        a.x = av.x;                                   // A[row][k+kh]
        a.y = av.y;                                   // A[row][k+kh+1]
        b.x = Bcol[(size_t)k * N];                    // B[k+kh][col]
        b.y = Bcol[(size_t)k * N + N];                // B[k+kh+1][col]
        acc = __builtin_amdgcn_wmma_f32_16x16x4_f32(
            /*neg_a=*/false, a, /*neg_b=*/false, b,
            /*c_mod=*/(short)0, acc, /*reuse_a=*/false, /*reuse_b=*/false);
    }

    const int   col   = n0 + lr;
    const float bv    = bias[col];
    const int   rbase = m0 + ((lane >> 4) << 3);      // +8 for upper half lanes
#pragma unroll
    for (int v = 0; v < 8; ++v) {
        float val = acc[v] + bv;
        if (relu) val = fmaxf(val, 0.0f);
        C[(size_t)(rbase + v) * N + col] = val;
    }
}

// ---------------------------------------------------------------------------
// Score pass (bandwidth-bound, streams conv_feat_embed once):
//   out[b, j] = mask[b, j] ? MIN_VALUE
//             : b_a + sum_c tanh(x[c] + h_embed[b, c]) * W_a[c]
// where x = fr_embed[b] for j == 0 else conv_feat_embed[b, j-1].
// One wave per (b, j) row; lane owns 16 contiguous channels -> b128 loads.
// The 268 MB conv_feat_embed stream exceeds the 192 MB L2, so those loads
// are non-temporal to keep h_embed / W_a resident; small reused operands
// use default (RT) policy.
// ---------------------------------------------------------------------------
__global__ void scores_kernel(const float* __restrict__ conv_embed, // [128,1023,512]
                              const float* __restrict__ fr_embed,   // [128,512]
                              const float* __restrict__ h_embed,    // [128,512]
                              const float* __restrict__ W_a,        // [512]
                              const float* __restrict__ b_a,        // [1]
                              const int*   __restrict__ mask,       // [128,1024]
                              float* __restrict__ out) {            // [128,1024]
    const int lane = threadIdx.x & 31;
    const int wave = threadIdx.x >> 5;
    const int row  = (blockIdx.x << 3) + wave;        // 0 .. 128*1024-1
    const int b    = row >> 10;
    const int j    = row & (COLS - 1);

    const float* x = (j == 0)
        ? (fr_embed + (size_t)b * ATT)
        : (conv_embed + ((size_t)b * ROI + (j - 1)) * ATT);
    const float* h = h_embed + (size_t)b * ATT;

    const int c0 = lane << 4;                         // 16 channels per lane
    float acc = 0.0f;
#pragma unroll
    for (int q = 0; q < 4; ++q) {
        const v4f xv = __builtin_nontemporal_load((const v4f*)(x + c0 + (q << 2)));
        const v4f hv = *(const v4f*)(h   + c0 + (q << 2));
        const v4f wv = *(const v4f*)(W_a + c0 + (q << 2));
        acc = __builtin_fmaf(tanhf(xv.x + hv.x), wv.x, acc);
        acc = __builtin_fmaf(tanhf(xv.y + hv.y), wv.y, acc);
        acc = __builtin_fmaf(tanhf(xv.z + hv.z), wv.z, acc);
        acc = __builtin_fmaf(tanhf(xv.w + hv.w), wv.w, acc);
    }
#pragma unroll
    for (int off = 16; off > 0; off >>= 1)
        acc += __shfl_xor(acc, off, 32);

    if (lane == 0) {
        const float s = acc + b_a[0];
        out[row] = mask[row] ? MIN_VALUE : s;
    }
}

extern "C" void kernel_launch(void* const* d_in, const int* in_sizes, int n_in,
                              void* d_out, int out_size, void* d_ws, size_t ws_size,
                              hipStream_t stream) {
    const float* h_out           = (const float*)d_in[0];   // [128,1024]
    const float* fake_region     = (const float*)d_in[1];   // [128,1024]
    /* d_in[2] = conv_feat: unused by the reference */
    const float* conv_feat_embed = (const float*)d_in[3];   // [128,1023,512]
    const int*   mask            = (const int*)  d_in[4];   // [128,1024]
    const float* W_f1            = (const float*)d_in[5];   // [1024,1024]
    const float* b_f1            = (const float*)d_in[6];   // [1024]
    const float* W_f2            = (const float*)d_in[7];   // [1024,512]
    const float* b_f2            = (const float*)d_in[8];   // [512]
    const float* W_h             = (const float*)d_in[9];   // [1024,512]
    const float* b_h             = (const float*)d_in[10];  // [512]
    const float* W_a             = (const float*)d_in[11];  // [512]
    const float* b_a             = (const float*)d_in[12];  // [1]
    float* out = (float*)d_out;

    float* ws     = (float*)d_ws;
    float* fr     = ws;                                  // 128*1024 f32 (512 KB)
    float* fr_emb = ws + (size_t)B_DIM * RNN;            // 128*512  f32 (256 KB)
    float* h_emb  = fr_emb + (size_t)B_DIM * ATT;        // 128*512  f32 (256 KB)

    // fr = relu(fake_region @ W_f1 + b_f1)
    gemm128_wmma_f32<<<RNN / 16, 256, 0, stream>>>(fake_region, W_f1, b_f1, fr,
                                                   RNN, RNN, 1);
    // fr_embed = fr @ W_f2 + b_f2
    gemm128_wmma_f32<<<ATT / 16, 256, 0, stream>>>(fr, W_f2, b_f2, fr_emb,
                                                   RNN, ATT, 0);
    // h_embed = h_out @ W_h + b_h
    gemm128_wmma_f32<<<ATT / 16, 256, 0, stream>>>(h_out, W_h, b_h, h_emb,
                                                   RNN, ATT, 0);
    // scores + mask (streams 268 MB of conv_feat_embed: the HBM-bound stage)
    scores_kernel<<<(B_DIM * COLS) / 8, 256, 0, stream>>>(
        conv_feat_embed, fr_emb, h_emb, W_a, b_a, mask, out);
}